// ImplicitLongConvolution_74380243632400
// MI455X (gfx1250) — compile-verified
//
#include <hip/hip_runtime.h>
#include <hip/hip_bf16.h>
#include <math.h>

typedef float v2f __attribute__((ext_vector_type(2)));
typedef float v8f __attribute__((ext_vector_type(8)));

#define LBLK 256
#define NWAVE 8

__device__ __forceinline__ v8f wmma4(v2f a, v2f b, v8f c) {
  // D = A(16x4,f32) * B(4x16,f32) + C(16x16,f32)
  return __builtin_amdgcn_wmma_f32_16x16x4_f32(false, a, false, b, (short)0, c,
                                               false, false);
}

__device__ __forceinline__ unsigned short f2bf(float f) {
  unsigned int u = __float_as_uint(f);
  u += 0x7FFFu + ((u >> 16) & 1u);   // round-to-nearest-even
  return (unsigned short)(u >> 16);
}
__device__ __forceinline__ float bf2f(unsigned short s) {
  return __uint_as_float(((unsigned int)s) << 16);
}

// ---------------------------------------------------------------------------
// 128x128 complex matmul C = A * B using V_WMMA_F32_16X16X4_F32.
// A is always complex (Ar_,Ai_). B may be real (Bi_ == nullptr) and may be
// logically transposed (bT). All pointers generic (LDS or global).
// 8 waves, each wave produces 8 of the 64 16x16 output tiles.
// f32 WMMA fragment layout (wave32):
//   A 16x4 : lane L: M=L%16, VGPR0 holds K=(L/16)*2, VGPR1 holds K+1
//   B 4x16 : lane L: N=L%16, VGPR0 holds K=(L/16)*2, VGPR1 holds K+1
//   C 16x16: VGPR r: lanes0-15 -> M=r, lanes16-31 -> M=r+8; N=L%16
// ---------------------------------------------------------------------------
__device__ void mm128(const float* __restrict__ Ar_, const float* __restrict__ Ai_,
                      const float* __restrict__ Br_, const float* __restrict__ Bi_,
                      bool bT, float* __restrict__ Cr_, float* __restrict__ Ci_) {
  const int tid  = threadIdx.x;
  const int wave = tid >> 5;
  const int lane = tid & 31;
  const int lm   = lane & 15;
  const int kb   = (lane >> 4) << 1;
  const int hi8  = (lane >> 4) << 3;
  for (int t = wave; t < 64; t += NWAVE) {
    const int tm = (t >> 3) << 4;
    const int tn = (t & 7) << 4;
    v8f cr = {0.f,0.f,0.f,0.f,0.f,0.f,0.f,0.f};
    v8f ci = {0.f,0.f,0.f,0.f,0.f,0.f,0.f,0.f};
    for (int kk = 0; kk < 128; kk += 4) {
      const int ka = kk + kb;
      v2f ar, ai, br;
      ar.x = Ar_[(tm + lm) * 128 + ka];
      ar.y = Ar_[(tm + lm) * 128 + ka + 1];
      ai.x = Ai_[(tm + lm) * 128 + ka];
      ai.y = Ai_[(tm + lm) * 128 + ka + 1];
      if (!bT) {
        br.x = Br_[ka * 128 + tn + lm];
        br.y = Br_[(ka + 1) * 128 + tn + lm];
      } else {
        br.x = Br_[(tn + lm) * 128 + ka];
        br.y = Br_[(tn + lm) * 128 + ka + 1];
      }
      cr = wmma4(ar, br, cr);
      ci = wmma4(ai, br, ci);
      if (Bi_) {
        v2f bi, nai;
        if (!bT) {
          bi.x = Bi_[ka * 128 + tn + lm];
          bi.y = Bi_[(ka + 1) * 128 + tn + lm];
        } else {
          bi.x = Bi_[(tn + lm) * 128 + ka];
          bi.y = Bi_[(tn + lm) * 128 + ka + 1];
        }
        nai.x = -ai.x;  // f32 WMMA has no A/B NEG modifier -> negate in VALU
        nai.y = -ai.y;
        cr = wmma4(nai, bi, cr);
        ci = wmma4(ar, bi, ci);
      }
    }
#pragma unroll
    for (int r = 0; r < 8; ++r) {
      Cr_[(tm + hi8 + r) * 128 + tn + lm] = cr[r];
      Ci_[(tm + hi8 + r) * 128 + tn + lm] = ci[r];
    }
  }
}

// ---------------------------------------------------------------------------
// Table init: F128 (forward + conj imag) and 16384-pt twiddles.
// Angles reduced exactly via (k*j) mod period before sin/cos.
// ---------------------------------------------------------------------------
__global__ void init_tables(float* __restrict__ Fr, float* __restrict__ Fi,
                            float* __restrict__ Fic,
                            float* __restrict__ Twr, float* __restrict__ Twi) {
  int idx = blockIdx.x * blockDim.x + threadIdx.x;
  if (idx >= 16384) return;
  int k = idx >> 7, j = idx & 127;
  const float TWO_PI = 6.28318530717958647692f;
  int p = (k * j) & 127;
  float a = -TWO_PI * (float)p * (1.0f / 128.0f);
  float s = sinf(a);
  Fr[idx]  = cosf(a);
  Fi[idx]  = s;
  Fic[idx] = -s;
  int pn = (k * j) & 16383;
  float a2 = -TWO_PI * (float)pn * (1.0f / 16384.0f);
  Twr[idx] = cosf(a2);
  Twi[idx] = sinf(a2);
}

// ---------------------------------------------------------------------------
// Coalesced 32x32 tiled transpose: in (R x C) -> out (C x R), per batch z.
// ---------------------------------------------------------------------------
__global__ void tpose32(const float* __restrict__ in, float* __restrict__ out,
                        int R, int C) {
  __shared__ float tile[32][33];
  const size_t base = (size_t)blockIdx.z * (size_t)R * (size_t)C;
  const int r0 = blockIdx.x * 32, c0 = blockIdx.y * 32;
  for (int j = threadIdx.y; j < 32; j += 8)
    tile[j][threadIdx.x] = in[base + (size_t)(r0 + j) * C + (c0 + threadIdx.x)];
  __syncthreads();
  for (int j = threadIdx.y; j < 32; j += 8)
    out[base + (size_t)(c0 + j) * R + (r0 + threadIdx.x)] = tile[threadIdx.x][j];
}

// ---------------------------------------------------------------------------
// Main fused kernel: one workgroup per channel d.
//   1) build k[:,d] via the tiny MLP (z = [t, cos(1e-4*w), -sin(1e-4*w)])
//   2) Kf = fourstep_FFT16384(k_pad); stash Kf/n as bf16 (64 KB LDS)
//   3) for b in {0,1}: Xf = FFT(x_pad); Z = Xf .* Kf/n; y = Re(FFTconj(Z^T))
// Four-step: stage1 B = F @ M; twiddle B[m][k] *= W_n^{mk}; stage2 A = B @ F.
// Inverse uses conj tables and reads Z transposed.
// LDS: Ar/Ai/Br/Bi (4 x 64 KB f32) + Kf bf16 (64 KB) = 320 KB (CDNA5 max).
// ---------------------------------------------------------------------------
__global__ void __launch_bounds__(256) conv_main(
    const float* __restrict__ xt,                       // (B, D, L)
    const float* __restrict__ W1, const float* __restrict__ b1,
    const float* __restrict__ W2, const float* __restrict__ b2,
    const float* __restrict__ Fr, const float* __restrict__ Fi,
    const float* __restrict__ Fic,
    const float* __restrict__ Twr, const float* __restrict__ Twi,
    float* __restrict__ yt)                             // (B, D, L)
{
  extern __shared__ float smem[];
  float* Ar  = smem;
  float* Ai  = smem + 16384;
  float* Brr = smem + 32768;
  float* Bii = smem + 49152;
  unsigned short* Kr = (unsigned short*)(smem + 65536);
  unsigned short* Ki = Kr + 16384;

  const int d   = blockIdx.x;
  const int tid = threadIdx.x;

  // ---- implicit filter k[:,d] (MLP over positional embedding), zero-padded
  for (int lin = tid; lin < 16384; lin += LBLK) {
    float v = 0.f;
    if (lin < 8192) {
      const float t0 = (float)lin * (1.0f / 8191.0f);
      const float ph = 1e-4f * (6.28318530717958647692f * (float)lin *
                                (1.0f / 8192.0f));
      const float z1 = cosf(ph);
      const float z2 = -sinf(ph);
      float acc = b2[d];
#pragma unroll
      for (int h = 0; h < 16; ++h) {
        float hv = t0 * W1[h] + z1 * W1[16 + h] + z2 * W1[32 + h] + b1[h];
        hv = fmaxf(hv, 0.f);
        acc += hv * W2[h * 1024 + d];
      }
      v = acc;
    }
    Ar[lin] = v;
  }
  __syncthreads();

  // ---- forward FFT of k (real input): B = F @ M ; twiddle ; A = B @ F
  mm128(Fr, Fi, Ar, nullptr, false, Brr, Bii);
  __syncthreads();
  for (int i = tid; i < 16384; i += LBLK) {
    float tr = Twr[i], ti = Twi[i];
    float br = Brr[i], bi = Bii[i];
    Brr[i] = br * tr - bi * ti;
    Bii[i] = br * ti + bi * tr;
  }
  __syncthreads();
  mm128(Brr, Bii, Fr, Fi, false, Ar, Ai);
  __syncthreads();
  for (int i = tid; i < 16384; i += LBLK) {   // stash Kf * (1/n) in bf16
    Kr[i] = f2bf(Ar[i] * (1.0f / 16384.0f));
    Ki[i] = f2bf(Ai[i] * (1.0f / 16384.0f));
  }
  __syncthreads();

  for (int b = 0; b < 2; ++b) {
    const float* xs = xt + ((size_t)b * 1024 + d) * 8192;
    for (int lin = tid; lin < 16384; lin += LBLK)
      Ar[lin] = (lin < 8192) ? xs[lin] : 0.f;
    __syncthreads();

    // forward FFT of x (real input)
    mm128(Fr, Fi, Ar, nullptr, false, Brr, Bii);
    __syncthreads();
    for (int i = tid; i < 16384; i += LBLK) {
      float tr = Twr[i], ti = Twi[i];
      float br = Brr[i], bi = Bii[i];
      Brr[i] = br * tr - bi * ti;
      Bii[i] = br * ti + bi * tr;
    }
    __syncthreads();
    mm128(Brr, Bii, Fr, Fi, false, Ar, Ai);
    __syncthreads();

    // pointwise: Z = Xf .* (Kf/n)
    for (int i = tid; i < 16384; i += LBLK) {
      float xr = Ar[i], xi = Ai[i];
      float kr = bf2f(Kr[i]), ki = bf2f(Ki[i]);
      Ar[i] = xr * kr - xi * ki;
      Ai[i] = xr * ki + xi * kr;
    }
    __syncthreads();

    // inverse FFT: conj twiddles, input read transposed (Z2^T)
    mm128(Fr, Fic, Ar, Ai, true, Brr, Bii);
    __syncthreads();
    for (int i = tid; i < 16384; i += LBLK) {
      float tr = Twr[i], ti = -Twi[i];
      float br = Brr[i], bi = Bii[i];
      Brr[i] = br * tr - bi * ti;
      Bii[i] = br * ti + bi * tr;
    }
    __syncthreads();
    mm128(Brr, Bii, Fr, Fic, false, Ar, Ai);
    __syncthreads();

    // y[j] = Re(A[j%128][j/128]), first L samples (causal conv result)
    float* ys = yt + ((size_t)b * 1024 + d) * 8192;
    for (int j = tid; j < 8192; j += LBLK)
      ys[j] = Ar[(j & 127) * 128 + (j >> 7)];
    __syncthreads();
  }
}

// ---------------------------------------------------------------------------
extern "C" void kernel_launch(void* const* d_in, const int* in_sizes, int n_in,
                              void* d_out, int out_size, void* d_ws, size_t ws_size,
                              hipStream_t stream) {
  const float* x  = (const float*)d_in[0];   // (2, 8192, 1024)
  const float* W1 = (const float*)d_in[1];   // (3, 16)
  const float* b1 = (const float*)d_in[2];   // (16,)
  const float* W2 = (const float*)d_in[3];   // (16, 1024)
  const float* b2 = (const float*)d_in[4];   // (1024,)
  float* out = (float*)d_out;                // (2, 8192, 1024)

  char* ws   = (char*)d_ws;
  float* Fr  = (float*)(ws);
  float* Fi  = (float*)(ws + 1 * 65536);
  float* Fic = (float*)(ws + 2 * 65536);
  float* Twr = (float*)(ws + 3 * 65536);
  float* Twi = (float*)(ws + 4 * 65536);
  float* xt  = (float*)(ws + (1u << 20));                              // 64 MB
  float* yt  = (float*)(ws + (1u << 20) + (size_t)64 * 1024 * 1024);   // 64 MB

  init_tables<<<64, 256, 0, stream>>>(Fr, Fi, Fic, Twr, Twi);

  dim3 tb(32, 8, 1);
  // x (B, L=8192, D=1024) -> xt (B, D, L): coalesced channel-major access
  tpose32<<<dim3(8192 / 32, 1024 / 32, 2), tb, 0, stream>>>(x, xt, 8192, 1024);

  const size_t smem = 4u * 65536u + 2u * 32768u;   // 320 KB (CDNA5 WGP max)
  conv_main<<<1024, 256, smem, stream>>>(xt, W1, b1, W2, b2,
                                         Fr, Fi, Fic, Twr, Twi, yt);

  // yt (B, D, L) -> out (B, L, D)
  tpose32<<<dim3(1024 / 32, 8192 / 32, 2), tb, 0, stream>>>(yt, out, 1024, 8192);
}